// LightGCN_81243601371549
// MI455X (gfx1250) — compile-verified
//
#include <hip/hip_runtime.h>

#define EMBED 128
#define WAVES_PER_BLOCK 8      // 256 threads = 8 wave32s
#define EDGES_PER_WAVE 128     // per-wave edge chunk staged via async-to-LDS

// ---------------------------------------------------------------------------
// init: hcur = concat(virus_emb, drug_emb); acc(d_out) = same; hnext = 0
// ---------------------------------------------------------------------------
__global__ void lgcn_init_kernel(const float* __restrict__ vemb,
                                 const float* __restrict__ demb,
                                 float* __restrict__ hcur,
                                 float* __restrict__ hnext,
                                 float* __restrict__ acc,
                                 long nv_elems, long n4) {
    long i = (long)blockIdx.x * blockDim.x + threadIdx.x;   // float4 index
    if (i >= n4) return;
    long e = i << 2;                                        // float index
    float4 x;
    if (e < nv_elems) {
        x = *(const float4*)(vemb + e);
    } else {
        x = *(const float4*)(demb + (e - nv_elems));
    }
    ((float4*)hcur)[i]  = x;
    ((float4*)acc)[i]   = x;
    ((float4*)hnext)[i] = make_float4(0.f, 0.f, 0.f, 0.f);
}

// ---------------------------------------------------------------------------
// SpMM over COO edges: hn[row] += val * h[col]
// - each wave32 owns a 128-edge chunk
// - edge triples staged into this wave's LDS slice with
//   global_load_async_to_lds_b128 (ASYNCcnt), consumed after s_wait_asynccnt
// - row/col forced uniform via readfirstlane -> scalar (SALU) base-address
//   math, saddr-form global_load_b128 gather and global_atomic_add_f32
//   scatter with a loop-invariant lane*16 voffset
// ---------------------------------------------------------------------------
__global__ void lgcn_spmm_kernel(const int*   __restrict__ erow,
                                 const int*   __restrict__ ecol,
                                 const float* __restrict__ eval,
                                 const float* __restrict__ h,
                                 float*       __restrict__ hn,
                                 int nedges) {
    __shared__ int   s_row[WAVES_PER_BLOCK][EDGES_PER_WAVE];
    __shared__ int   s_col[WAVES_PER_BLOCK][EDGES_PER_WAVE];
    __shared__ float s_val[WAVES_PER_BLOCK][EDGES_PER_WAVE];

    const int wid  = (int)(threadIdx.x >> 5);
    const int lane = (int)(threadIdx.x & 31);

    long base = ((long)blockIdx.x * WAVES_PER_BLOCK + wid) * EDGES_PER_WAVE;
    if (base >= nedges) return;                 // wave-uniform exit
    long rem   = (long)nedges - base;
    int  count = rem < EDGES_PER_WAVE ? (int)rem : EDGES_PER_WAVE;

    if (count == EDGES_PER_WAVE) {
        // ---- async stage: 3 x 512B (one b128 per lane per array) ----
        unsigned goff  = (unsigned)(base * 4) + (unsigned)(lane * 16);
        unsigned lds_r = (unsigned)(uintptr_t)&s_row[wid][0] + (unsigned)(lane * 16);
        unsigned lds_c = (unsigned)(uintptr_t)&s_col[wid][0] + (unsigned)(lane * 16);
        unsigned lds_v = (unsigned)(uintptr_t)&s_val[wid][0] + (unsigned)(lane * 16);
        asm volatile("global_load_async_to_lds_b128 %0, %1, %2"
                     :: "v"(lds_r), "v"(goff), "s"(erow) : "memory");
        asm volatile("global_load_async_to_lds_b128 %0, %1, %2"
                     :: "v"(lds_c), "v"(goff), "s"(ecol) : "memory");
        asm volatile("global_load_async_to_lds_b128 %0, %1, %2"
                     :: "v"(lds_v), "v"(goff), "s"(eval) : "memory");
        asm volatile("s_wait_asynccnt 0x0" ::: "memory");
    } else {
        // ---- tail chunk: cooperative direct fill ----
        for (int k = lane; k < count; k += 32) {
            s_row[wid][k] = erow[base + k];
            s_col[wid][k] = ecol[base + k];
            s_val[wid][k] = eval[base + k];
        }
        // wave-private data; same-wave LDS ordering + compiler dscnt waits
    }

    const size_t lane_off = (size_t)lane * 4;   // loop-invariant voffset

    for (int k = 0; k < count; ++k) {
        // all lanes read the same LDS word -> uniform; pin into SGPRs so the
        // backend does base-address math on the scalar unit and uses
        // saddr-form vector memory ops
        int   row = __builtin_amdgcn_readfirstlane(s_row[wid][k]);
        int   col = __builtin_amdgcn_readfirstlane(s_col[wid][k]);
        float val = s_val[wid][k];

        const float* __restrict__ src = h  + (size_t)col * EMBED;  // scalar base
        float*       __restrict__ dst = hn + (size_t)row * EMBED;  // scalar base

        float4 m = *(const float4*)(src + lane_off);

        unsafeAtomicAdd(dst + lane_off + 0, m.x * val);  // global_atomic_add_f32 (L2)
        unsafeAtomicAdd(dst + lane_off + 1, m.y * val);
        unsafeAtomicAdd(dst + lane_off + 2, m.z * val);
        unsafeAtomicAdd(dst + lane_off + 3, m.w * val);
    }
}

// ---------------------------------------------------------------------------
// acc += hn ; zero the buffer that becomes the next scatter target
// ---------------------------------------------------------------------------
__global__ void lgcn_accum_zero_kernel(float* __restrict__ acc,
                                       const float* __restrict__ hn,
                                       float* __restrict__ zbuf,
                                       long n4) {
    long i = (long)blockIdx.x * blockDim.x + threadIdx.x;
    if (i >= n4) return;
    float4 a = ((const float4*)acc)[i];
    float4 b = ((const float4*)hn)[i];
    a.x += b.x; a.y += b.y; a.z += b.z; a.w += b.w;
    ((float4*)acc)[i]  = a;
    ((float4*)zbuf)[i] = make_float4(0.f, 0.f, 0.f, 0.f);
}

// ---------------------------------------------------------------------------
// out = (acc + hn) * 0.25   (mean over L+1 = 4 layers)
// ---------------------------------------------------------------------------
__global__ void lgcn_final_kernel(float* __restrict__ acc,
                                  const float* __restrict__ hn,
                                  long n4) {
    long i = (long)blockIdx.x * blockDim.x + threadIdx.x;
    if (i >= n4) return;
    float4 a = ((const float4*)acc)[i];
    float4 b = ((const float4*)hn)[i];
    a.x = (a.x + b.x) * 0.25f;
    a.y = (a.y + b.y) * 0.25f;
    a.z = (a.z + b.z) * 0.25f;
    a.w = (a.w + b.w) * 0.25f;
    ((float4*)acc)[i] = a;
}

// ---------------------------------------------------------------------------
extern "C" void kernel_launch(void* const* d_in, const int* in_sizes, int n_in,
                              void* d_out, int out_size, void* d_ws, size_t ws_size,
                              hipStream_t stream) {
    const int*   erow = (const int*)  d_in[0];
    const int*   ecol = (const int*)  d_in[1];
    const float* eval = (const float*)d_in[2];
    const float* vemb = (const float*)d_in[3];
    const float* demb = (const float*)d_in[4];

    const long nedges   = in_sizes[0];                            // 1.6M edges
    const long nv_elems = in_sizes[3];                            // 50000*128
    const long total    = (long)in_sizes[3] + (long)in_sizes[4];  // N*128 floats
    const long n4       = total >> 2;                             // float4 count

    float* acc = (float*)d_out;            // running layer sum lives in d_out
    float* h0  = (float*)d_ws;             // ping
    float* h1  = h0 + total;               // pong (2 x 51.2MB in d_ws)

    const int  TPB = 256;
    const long eb  = (n4 + TPB - 1) / TPB;
    lgcn_init_kernel<<<dim3((unsigned)eb), dim3(TPB), 0, stream>>>(
        vemb, demb, h0, h1, acc, nv_elems, n4);

    float* hc = h0;
    float* hn = h1;
    const long edges_per_block = (long)WAVES_PER_BLOCK * EDGES_PER_WAVE;
    const long spmm_blocks = (nedges + edges_per_block - 1) / edges_per_block;

    for (int layer = 0; layer < 3; ++layer) {
        lgcn_spmm_kernel<<<dim3((unsigned)spmm_blocks), dim3(TPB), 0, stream>>>(
            erow, ecol, eval, hc, hn, (int)nedges);
        if (layer < 2) {
            lgcn_accum_zero_kernel<<<dim3((unsigned)eb), dim3(TPB), 0, stream>>>(
                acc, hn, hc, n4);
        } else {
            lgcn_final_kernel<<<dim3((unsigned)eb), dim3(TPB), 0, stream>>>(
                acc, hn, n4);
        }
        float* t = hc; hc = hn; hn = t;
    }
}